// Prior_SPDE_20203526160573
// MI455X (gfx1250) — compile-verified
//
#include <hip/hip_runtime.h>
#include <hip/hip_bf16.h>

// Problem constants (fixed by setup_inputs): B=2, N=1024 (32x32 grid), T=5,
// dt=dx=dy=1. Output Q: (B, T*N, T*N) f32 block-tridiagonal.
#define BB   2
#define NN   1024
#define NX   32
#define TT   5
#define NT   (TT * NN)          // 5120
#define BT   (BB * TT)          // 10 (b,t) operators

typedef __attribute__((ext_vector_type(2))) float v2f;
typedef __attribute__((ext_vector_type(8))) float v8f;

// ---------------------------------------------------------------------------
// K0: float4 zero fill (used for d_out and dense-M workspace)
// ---------------------------------------------------------------------------
__global__ void k_zero4(float4* __restrict__ p, size_t n4) {
    size_t i = (size_t)blockIdx.x * blockDim.x + threadIdx.x;
    size_t stride = (size_t)gridDim.x * blockDim.x;
    float4 z = make_float4(0.f, 0.f, 0.f, 0.f);
    for (; i < n4; i += stride) p[i] = z;
}

// ---------------------------------------------------------------------------
// K1: per-row 9-point stencil coefficients of A, and d = 1/tau^2.
// Storage order o = (dy+1)*3 + (dx+1); invalid (clipped) offsets stored as 0.
// tid = (b*T + t)*N + n
// ---------------------------------------------------------------------------
__global__ void k_coeff(const float* __restrict__ kappa, const float* __restrict__ m,
                        const float* __restrict__ H, const float* __restrict__ tau,
                        float* __restrict__ Ac, float* __restrict__ dv) {
    int tid = blockIdx.x * blockDim.x + threadIdx.x;
    if (tid >= BT * NN) return;
    int n = tid % NN;
    int t = (tid / NN) % TT;
    int b = tid / (NN * TT);

    size_t sNT = (size_t)NN * TT;
    size_t e = (size_t)n * TT + t;
    float kp  = kappa[(size_t)b * sNT + e];
    float m1  = m[((size_t)b * 2 + 0) * sNT + e];
    float m2  = m[((size_t)b * 2 + 1) * sNT + e];
    float H11 = H[(((size_t)b * 2 + 0) * 2 + 0) * sNT + e];
    float H12 = H[(((size_t)b * 2 + 0) * 2 + 1) * sNT + e];
    float H21 = H[(((size_t)b * 2 + 1) * 2 + 0) * sNT + e];
    float H22 = H[(((size_t)b * 2 + 1) * 2 + 1) * sNT + e];
    float ta  = tau[(size_t)b * sNT + e];

    float k2 = kp * kp;
    float hs = H12 + H21;
    int x = n & (NX - 1), y = n >> 5;

    float a[9];
    a[4] = k2 + 2.f * H11 + 2.f * H22;                       // ( 0, 0)
    a[3] = (x >= 1)              ? (-0.5f * m1 - H11) : 0.f; // ( 0,-1)
    a[5] = (x <= NX - 2)         ? ( 0.5f * m1 - H11) : 0.f; // ( 0,+1)
    a[1] = (y >= 1)              ? (-0.5f * m2 - H22) : 0.f; // (-1, 0)
    a[7] = (y <= NX - 2)         ? ( 0.5f * m2 - H22) : 0.f; // (+1, 0)
    a[0] = (x >= 1 && y >= 1)    ? (-0.25f * hs) : 0.f;      // (-1,-1)
    a[2] = (x <= NX-2 && y >= 1) ? ( 0.25f * hs) : 0.f;      // (-1,+1)
    a[6] = (x >= 1 && y <= NX-2) ? ( 0.25f * hs) : 0.f;      // (+1,-1)
    a[8] = (x <= NX-2 && y <= NX-2) ? (-0.25f * hs) : 0.f;   // (+1,+1)

    float* dst = Ac + (size_t)tid * 9;
#pragma unroll
    for (int o = 0; o < 9; ++o) dst[o] = a[o];
    dv[tid] = 1.0f / (ta * ta);
}

// ---------------------------------------------------------------------------
// K2: M = I + A*A as a 25-point stencil per row (offsets dy,dx in [-2,2]).
// ---------------------------------------------------------------------------
__global__ void k_m25(const float* __restrict__ Ac, float* __restrict__ m25) {
    int tid = blockIdx.x * blockDim.x + threadIdx.x;
    if (tid >= BT * NN) return;
    int n = tid % NN;
    int x = n & (NX - 1), y = n >> 5;

    float acc[25];
#pragma unroll
    for (int i = 0; i < 25; ++i) acc[i] = 0.f;

    const float* arow = Ac + (size_t)tid * 9;
#pragma unroll
    for (int o1 = 0; o1 < 9; ++o1) {
        int sy1 = o1 / 3 - 1, sx1 = o1 % 3 - 1;
        int x1 = x + sx1, y1 = y + sy1;
        if (x1 < 0 || x1 > NX - 1 || y1 < 0 || y1 > NX - 1) continue;
        float a1 = arow[o1];
        const float* arow2 = Ac + (size_t)(tid + sy1 * NX + sx1) * 9;
#pragma unroll
        for (int o2 = 0; o2 < 9; ++o2) {
            int sy2 = o2 / 3 - 1, sx2 = o2 % 3 - 1;
            acc[(sy1 + sy2 + 2) * 5 + (sx1 + sx2 + 2)] += a1 * arow2[o2];
        }
    }
    acc[12] += 1.0f;  // + I (dt = 1)

    float* dst = m25 + (size_t)tid * 25;
#pragma unroll
    for (int i = 0; i < 25; ++i) dst[i] = acc[i];
}

// ---------------------------------------------------------------------------
// K3: scatter the 25-point stencil into dense N x N M (pre-zeroed).
// ---------------------------------------------------------------------------
__global__ void k_scatter(const float* __restrict__ m25, float* __restrict__ Md) {
    int tid = blockIdx.x * blockDim.x + threadIdx.x;
    if (tid >= BT * NN) return;
    int n = tid % NN;
    int x = n & (NX - 1), y = n >> 5;
    size_t bt = (size_t)tid / NN;
    const float* src = m25 + (size_t)tid * 25;
    float* row = Md + bt * NN * NN + (size_t)n * NN;
#pragma unroll
    for (int o = 0; o < 25; ++o) {
        int dy = o / 5 - 2, dx = o % 5 - 2;
        int xx = x + dx, yy = y + dy;
        if (xx < 0 || xx > NX - 1 || yy < 0 || yy > NX - 1) continue;
        row[n + dy * NX + dx] = src[o];
    }
}

// ---------------------------------------------------------------------------
// K4: off-diagonal blocks (output pre-zeroed):
//   super (j-1, j): (r=n, c=n+off)      val = -d_j[n] * M_j[n, n+off]
//   sub   (j, j-1): (r=n+off, c=n)      same value (transpose)
// ---------------------------------------------------------------------------
__global__ void k_offdiag(const float* __restrict__ m25, const float* __restrict__ dv,
                          float* __restrict__ out) {
    int tid = blockIdx.x * blockDim.x + threadIdx.x;
    if (tid >= BB * 4 * NN) return;
    int n = tid % NN;
    int j = (tid / NN) % 4 + 1;   // time index 1..4
    int b = tid / (4 * NN);
    int x = n & (NX - 1), y = n >> 5;

    size_t btj = (size_t)(b * TT + j) * NN;
    float dj = dv[btj + n];
    const float* src = m25 + (btj + n) * 25;
    size_t obase = (size_t)b * NT * NT;
#pragma unroll
    for (int o = 0; o < 25; ++o) {
        int dy = o / 5 - 2, dx = o % 5 - 2;
        int xx = x + dx, yy = y + dy;
        if (xx < 0 || xx > NX - 1 || yy < 0 || yy > NX - 1) continue;
        int c = n + dy * NX + dx;
        float v = -dj * src[o];
        out[obase + (size_t)((j - 1) * NN + n) * NT + (size_t)j * NN + c] = v;  // super
        out[obase + (size_t)(j * NN + c) * NT + (size_t)(j - 1) * NN + n] = v;  // sub
    }
}

// ---------------------------------------------------------------------------
// K5: diagonal blocks S_i = M_i^T diag(d_i) M_i (+ d_{i+1} I for i < T-1)
// Band-limited WMMA f32 16x16x4 GEMM. M has band half-width 66, so only tiles
// with |tp - tq| <= 9 are nonzero; K restricted to the band overlap.
// One wave (wave32) per 16x16 output tile; EXEC is full through the K-loop.
// ---------------------------------------------------------------------------
#define BAND_T 19   // tq - tp in [-9, 9]
#define TILES_PER_BLK (64 * BAND_T)

__global__ void __launch_bounds__(128) k_diag_wmma(const float* __restrict__ Md,
                                                   const float* __restrict__ dv,
                                                   float* __restrict__ out) {
    int wave = blockIdx.x * (blockDim.x >> 5) + (threadIdx.x >> 5);
    if (wave >= BT * TILES_PER_BLK) return;
    int bt = wave / TILES_PER_BLK;
    int rem = wave % TILES_PER_BLK;
    int tp = rem / BAND_T;
    int tq = tp + (rem % BAND_T) - 9;
    if (tq < 0 || tq >= 64) return;            // wave-uniform exit
    int b = bt / TT, i = bt % TT;

    int p0 = tp * 16, q0 = tq * 16;
    int hiP = (p0 > q0) ? p0 : q0;
    int loP = (p0 < q0) ? p0 : q0;
    int klo = hiP - 66; if (klo < 0) klo = 0; klo &= ~3;
    int khi = loP + 82; khi = (khi + 3) & ~3; if (khi > NN) khi = NN;

    const float* __restrict__ Mt  = Md + (size_t)bt * NN * NN;
    const float* __restrict__ dvt = dv + (size_t)bt * NN;

    int lane = threadIdx.x & 31;
    int half = lane >> 4;        // 0: K=0,1  1: K=2,3
    int l15  = lane & 15;        // matrix row (A) / col (B)

    v8f acc = {0.f, 0.f, 0.f, 0.f, 0.f, 0.f, 0.f, 0.f};

    for (int n0 = klo; n0 < khi; n0 += 4) {
        int na = n0 + 2 * half;
        const float* r0 = Mt + (size_t)na * NN;
        const float* r1 = r0 + NN;
        // prefetch next K-step rows (global_prefetch_b8)
        __builtin_prefetch(r0 + 4 * NN + p0 + l15, 0, 0);
        __builtin_prefetch(r0 + 4 * NN + q0 + l15, 0, 0);
        float da = dvt[na], db = dvt[na + 1];
        v2f a, bv;
        a[0]  = r0[p0 + l15] * da;   // A = (M^T diag(d))[p, k]
        a[1]  = r1[p0 + l15] * db;
        bv[0] = r0[q0 + l15];        // B = M[k, q]
        bv[1] = r1[q0 + l15];
        acc = __builtin_amdgcn_wmma_f32_16x16x4_f32(
            /*neg_a=*/false, a, /*neg_b=*/false, bv,
            /*c_mod=*/(short)0, acc, /*reuse_a=*/false, /*reuse_b=*/false);
    }

    // C/D layout: vgpr v, lanes 0-15 -> row v, lanes 16-31 -> row v+8; col = l15
    size_t obase = (size_t)b * NT * NT + (size_t)i * NN * NT + (size_t)i * NN;
#pragma unroll
    for (int v = 0; v < 8; ++v) {
        int row = v + 8 * half;
        float val = acc[v];
        if (i < TT - 1 && tp == tq && l15 == row)
            val += dv[(size_t)(b * TT + i + 1) * NN + p0 + row];  // + d_{i+1} I
        out[obase + (size_t)(p0 + row) * NT + q0 + l15] = val;
    }
}

// ---------------------------------------------------------------------------
// Host launcher. Workspace layout (floats), requires ~43.4 MB:
//   Md   : BT * N * N       (dense M, zero-filled then scattered)
//   m25  : BT * N * 25
//   Ac   : BT * N * 9
//   dv   : BT * N
// ---------------------------------------------------------------------------
extern "C" void kernel_launch(void* const* d_in, const int* in_sizes, int n_in,
                              void* d_out, int out_size, void* d_ws, size_t ws_size,
                              hipStream_t stream) {
    const float* kappa = (const float*)d_in[0];
    const float* m     = (const float*)d_in[1];
    const float* H     = (const float*)d_in[2];
    const float* tau   = (const float*)d_in[3];
    float* out = (float*)d_out;

    float* Md  = (float*)d_ws;
    float* m25 = Md  + (size_t)BT * NN * NN;
    float* Ac  = m25 + (size_t)BT * NN * 25;
    float* dv  = Ac  + (size_t)BT * NN * 9;

    // Zero output (210 MB) and dense M (40 MB)
    {
        size_t n4 = (size_t)BB * NT * NT / 4;
        k_zero4<<<4096, 256, 0, stream>>>((float4*)out, n4);
        size_t m4 = (size_t)BT * NN * NN / 4;
        k_zero4<<<4096, 256, 0, stream>>>((float4*)Md, m4);
    }

    int nRows = BT * NN;  // 10240
    k_coeff<<<(nRows + 255) / 256, 256, 0, stream>>>(kappa, m, H, tau, Ac, dv);
    k_m25<<<(nRows + 255) / 256, 256, 0, stream>>>(Ac, m25);
    k_scatter<<<(nRows + 255) / 256, 256, 0, stream>>>(m25, Md);

    int nOff = BB * 4 * NN;  // 8192
    k_offdiag<<<(nOff + 255) / 256, 256, 0, stream>>>(m25, dv, out);

    int nWaves = BT * TILES_PER_BLK;           // 12160 tiles, 1 wave each
    int nBlks = (nWaves + 3) / 4;              // 4 waves (128 threads) per block
    k_diag_wmma<<<nBlks, 128, 0, stream>>>(Md, dv, out);
}